// NodeSinkhornPooling_36266703847506
// MI455X (gfx1250) — compile-verified
//
#include <hip/hip_runtime.h>

typedef __attribute__((ext_vector_type(2))) float v2f;
typedef __attribute__((ext_vector_type(8))) float v8f;

#define S_DIM   64
#define K_DIM   64
#define D_DIM   128
#define LDC     65          // stride-65 padding: conflict-free row AND column scans
#define NITERS  50

// eps = 0.3 ; everything in base-2 log domain:
//   Ce = C / (eps*ln2), fe = f/(eps*ln2), ge = g/(eps*ln2)
//   fe_i = -(LSE2_j(ge_j - Ce_ij) - 6)       (log2 K = 6)
//   hist_j = 2^( LSE2_i(fe_i - Ce_ij) + ge_j - 12 ),  then normalize
#define INV_EPS_LN2 4.80898346962988f   // 1/(0.3*ln2)

__device__ __forceinline__ float ex2(float x) { return __builtin_amdgcn_exp2f(x); }
__device__ __forceinline__ float lg2(float x) { return __builtin_amdgcn_logf(x); }

__global__ __launch_bounds__(64)
void node_sinkhorn_kernel(const float* __restrict__ samples,
                          const float* __restrict__ codebook,
                          float* __restrict__ out)
{
    __shared__ float Ce[S_DIM * LDC];
    __shared__ float fe[S_DIM], ge[K_DIM], sqx[S_DIM], sqy[K_DIM], hh[K_DIM];

    const int node = blockIdx.x;
    const int tid  = threadIdx.x;          // 0..63
    const float* __restrict__ X = samples + (size_t)node * S_DIM * D_DIM;
    const float* __restrict__ Y = codebook;

    // ---------------- phase 0: squared norms (one row per thread) ----------
    {
        const float4* xr = (const float4*)(X + tid * D_DIM);
        const float4* yr = (const float4*)(Y + tid * D_DIM);
        float ax = 0.f, ay = 0.f;
        #pragma unroll 8
        for (int q = 0; q < D_DIM / 4; ++q) {
            float4 v = xr[q]; ax += v.x*v.x + v.y*v.y + v.z*v.z + v.w*v.w;
            float4 u = yr[q]; ay += u.x*u.x + u.y*u.y + u.z*u.z + u.w*u.w;
        }
        sqx[tid] = ax;
        sqy[tid] = ay;
    }
    __syncthreads();

    // ---------------- phase 1: XY^T via V_WMMA_F32_16X16X4_F32 -------------
    // wave w owns rows [32w, 32w+32); tiles: 2 row-strips x 4 col-tiles.
    const int wave = tid >> 5;             // 0..1
    const int lane = tid & 31;
    const int half = lane >> 4;            // 0: K=k0..k0+1, 1: K=k0+2..k0+3
    const int r    = lane & 15;

    v8f acc[2][4];
    #pragma unroll
    for (int s = 0; s < 2; ++s)
        #pragma unroll
        for (int c = 0; c < 4; ++c)
            acc[s][c] = (v8f){};

    #pragma unroll 4
    for (int k0 = 0; k0 < D_DIM; k0 += 4) {
        v2f a[2], b[4];
        #pragma unroll
        for (int s = 0; s < 2; ++s)
            a[s] = *(const v2f*)(X + (wave*32 + s*16 + r) * D_DIM + k0 + 2*half);
        #pragma unroll
        for (int c = 0; c < 4; ++c)
            b[c] = *(const v2f*)(Y + (c*16 + r) * D_DIM + k0 + 2*half);
        #pragma unroll
        for (int s = 0; s < 2; ++s)
            #pragma unroll
            for (int c = 0; c < 4; ++c)
                acc[s][c] = __builtin_amdgcn_wmma_f32_16x16x4_f32(
                    false, a[s], false, b[c], (short)0, acc[s][c], false, false);
    }

    // cost -> LDS (already divided by eps*ln2); clamp at 0 like the reference
    #pragma unroll
    for (int s = 0; s < 2; ++s)
        #pragma unroll
        for (int c = 0; c < 4; ++c)
            #pragma unroll
            for (int e = 0; e < 8; ++e) {
                int M = wave*32 + s*16 + 8*half + e;
                int Nn = c*16 + r;
                float cv = sqx[M] + sqy[Nn] - 2.0f * acc[s][c][e];
                Ce[M * LDC + Nn] = fmaxf(cv, 0.0f) * INV_EPS_LN2;
            }

    fe[tid] = 0.0f;
    ge[tid] = 0.0f;
    __syncthreads();

    // ---------------- phase 2: log-domain Sinkhorn (base-2) ----------------
    for (int it = 0; it < NITERS; ++it) {
        // f-update: thread = row i, scan columns (LDS conflict-free)
        {
            const float* row = &Ce[tid * LDC];
            float m0 = -INFINITY, m1 = -INFINITY, m2 = -INFINITY, m3 = -INFINITY;
            #pragma unroll 4
            for (int j = 0; j < K_DIM; j += 4) {
                m0 = fmaxf(m0, ge[j+0] - row[j+0]);
                m1 = fmaxf(m1, ge[j+1] - row[j+1]);
                m2 = fmaxf(m2, ge[j+2] - row[j+2]);
                m3 = fmaxf(m3, ge[j+3] - row[j+3]);
            }
            float m = fmaxf(fmaxf(m0, m1), fmaxf(m2, m3));
            float s0 = 0.f, s1 = 0.f, s2 = 0.f, s3 = 0.f;
            #pragma unroll 4
            for (int j = 0; j < K_DIM; j += 4) {
                s0 += ex2(ge[j+0] - row[j+0] - m);
                s1 += ex2(ge[j+1] - row[j+1] - m);
                s2 += ex2(ge[j+2] - row[j+2] - m);
                s3 += ex2(ge[j+3] - row[j+3] - m);
            }
            fe[tid] = -(m + lg2((s0 + s1) + (s2 + s3)) - 6.0f);
        }
        __syncthreads();
        // g-update: thread = column j, scan rows (stride-65 => conflict-free)
        {
            const float* col = &Ce[tid];
            float m0 = -INFINITY, m1 = -INFINITY, m2 = -INFINITY, m3 = -INFINITY;
            #pragma unroll 4
            for (int i = 0; i < S_DIM; i += 4) {
                m0 = fmaxf(m0, fe[i+0] - col[(i+0)*LDC]);
                m1 = fmaxf(m1, fe[i+1] - col[(i+1)*LDC]);
                m2 = fmaxf(m2, fe[i+2] - col[(i+2)*LDC]);
                m3 = fmaxf(m3, fe[i+3] - col[(i+3)*LDC]);
            }
            float m = fmaxf(fmaxf(m0, m1), fmaxf(m2, m3));
            float s0 = 0.f, s1 = 0.f, s2 = 0.f, s3 = 0.f;
            #pragma unroll 4
            for (int i = 0; i < S_DIM; i += 4) {
                s0 += ex2(fe[i+0] - col[(i+0)*LDC] - m);
                s1 += ex2(fe[i+1] - col[(i+1)*LDC] - m);
                s2 += ex2(fe[i+2] - col[(i+2)*LDC] - m);
                s3 += ex2(fe[i+3] - col[(i+3)*LDC] - m);
            }
            ge[tid] = -(m + lg2((s0 + s1) + (s2 + s3)) - 6.0f);
        }
        __syncthreads();
    }

    // ---------------- phase 3: column marginal histogram -------------------
    {
        const float* col = &Ce[tid];
        float m = -INFINITY;
        #pragma unroll 4
        for (int i = 0; i < S_DIM; ++i)
            m = fmaxf(m, fe[i] - col[i * LDC]);
        float s0 = 0.f, s1 = 0.f, s2 = 0.f, s3 = 0.f;
        #pragma unroll 4
        for (int i = 0; i < S_DIM; i += 4) {
            s0 += ex2(fe[i+0] - col[(i+0)*LDC] - m);
            s1 += ex2(fe[i+1] - col[(i+1)*LDC] - m);
            s2 += ex2(fe[i+2] - col[(i+2)*LDC] - m);
            s3 += ex2(fe[i+3] - col[(i+3)*LDC] - m);
        }
        // log2(pi) column-LSE + ge_j + (log_a+log_b)/ln2 = ... - 12
        float h = ex2(m + lg2((s0 + s1) + (s2 + s3)) + ge[tid] - 12.0f);
        hh[tid] = h;
        __syncthreads();
        float tot = 0.f;
        #pragma unroll 8
        for (int j = 0; j < K_DIM; ++j) tot += hh[j];
        out[(size_t)node * K_DIM + tid] = h / (tot + 1e-12f);
    }
}

extern "C" void kernel_launch(void* const* d_in, const int* in_sizes, int n_in,
                              void* d_out, int out_size, void* d_ws, size_t ws_size,
                              hipStream_t stream)
{
    const float* samples  = (const float*)d_in[0];   // (8192, 64, 128) f32
    const float* codebook = (const float*)d_in[1];   // (64, 128) f32
    float* out = (float*)d_out;                      // (8192, 64) f32

    const int N = in_sizes[0] / (S_DIM * D_DIM);     // 8192 nodes
    node_sinkhorn_kernel<<<dim3(N), dim3(64), 0, stream>>>(samples, codebook, out);
}